// PGNN_5634997092468
// MI455X (gfx1250) — compile-verified
//
#include <hip/hip_runtime.h>
#include <hip/hip_bf16.h>

typedef float v2f __attribute__((ext_vector_type(2)));
typedef float v8f __attribute__((ext_vector_type(8)));

#define NNODES 50000
#define NANCH 64
#define IN_DIM 128
#define FEATD 32

// ---------------------------------------------------------------------------
// Kernel 1: x = feat @ W_pre + b_pre     [N,128] x [128,32] -> [N,32]
// One wave handles a 16-node tile, producing 16x32 outputs via two 16x16 C
// tiles, K swept in steps of 4 with V_WMMA_F32_16X16X4_F32 (exact fp32).
// A layout (16x4 f32): lane m = lane&15; VGPR-half (lane>>4) picks K pair.
// B layout (4x16 f32): VGPR i holds rows K=i (lanes 0-15) and K=i+2 (16-31).
// ---------------------------------------------------------------------------
__global__ void pgnn_pre_gemm(const float* __restrict__ feat,
                              const float* __restrict__ W,
                              const float* __restrict__ b,
                              float* __restrict__ x, int n_tiles) {
  int wid = blockIdx.x * (blockDim.x >> 5) + (threadIdx.x >> 5);
  if (wid >= n_tiles) return;                 // wave-uniform: EXEC stays all-1s
  int lane = threadIdx.x & 31;
  int ln = lane & 15;                         // M index for A, N index for B/C
  int kh = lane >> 4;                         // selects K pair {0,1} vs {2,3}
  int row0 = wid * 16;

  v8f acc0, acc1;
  float b0 = b[ln], b1 = b[16 + ln];
  #pragma unroll
  for (int i = 0; i < 8; ++i) { acc0[i] = b0; acc1[i] = b1; }

  for (int k0 = 0; k0 < IN_DIM; k0 += 4) {
    int ka = k0 + 2 * kh;
    v2f a;
    a.x = feat[(row0 + ln) * IN_DIM + ka];
    a.y = feat[(row0 + ln) * IN_DIM + ka + 1];
    v2f wb0, wb1;
    wb0.x = W[ka * FEATD + ln];        wb0.y = W[(ka + 1) * FEATD + ln];
    wb1.x = W[ka * FEATD + 16 + ln];   wb1.y = W[(ka + 1) * FEATD + 16 + ln];
    acc0 = __builtin_amdgcn_wmma_f32_16x16x4_f32(false, a, false, wb0,
                                                 (short)0, acc0, false, false);
    acc1 = __builtin_amdgcn_wmma_f32_16x16x4_f32(false, a, false, wb1,
                                                 (short)0, acc1, false, false);
  }
  #pragma unroll
  for (int i = 0; i < 8; ++i) {
    int m = row0 + i + 8 * kh;                // C VGPR i: rows i and i+8
    x[m * FEATD + ln]      = acc0[i];
    x[m * FEATD + 16 + ln] = acc1[i];
  }
}

// ---------------------------------------------------------------------------
// Kernel 2: u = x@Wu + bu ; v = x@Wv + bv    [N,32] x [32,32], two outputs.
// Same wave-tiling as above; the A tile (16x4) is loaded once per K-step and
// shared across the 4 accumulators (u-lo, u-hi, v-lo, v-hi).
// ---------------------------------------------------------------------------
__global__ void pgnn_uv_gemm(const float* __restrict__ x,
                             const float* __restrict__ Wu, const float* __restrict__ bu,
                             const float* __restrict__ Wv, const float* __restrict__ bv,
                             float* __restrict__ u, float* __restrict__ v,
                             int n_tiles) {
  int wid = blockIdx.x * (blockDim.x >> 5) + (threadIdx.x >> 5);
  if (wid >= n_tiles) return;
  int lane = threadIdx.x & 31;
  int ln = lane & 15;
  int kh = lane >> 4;
  int row0 = wid * 16;

  v8f au0, au1, av0, av1;
  float bu0 = bu[ln], bu1 = bu[16 + ln];
  float bv0 = bv[ln], bv1 = bv[16 + ln];
  #pragma unroll
  for (int i = 0; i < 8; ++i) { au0[i] = bu0; au1[i] = bu1; av0[i] = bv0; av1[i] = bv1; }

  #pragma unroll
  for (int k0 = 0; k0 < FEATD; k0 += 4) {
    int ka = k0 + 2 * kh;
    v2f a;
    a.x = x[(row0 + ln) * FEATD + ka];
    a.y = x[(row0 + ln) * FEATD + ka + 1];
    v2f wu0, wu1, wv0, wv1;
    wu0.x = Wu[ka * FEATD + ln];        wu0.y = Wu[(ka + 1) * FEATD + ln];
    wu1.x = Wu[ka * FEATD + 16 + ln];   wu1.y = Wu[(ka + 1) * FEATD + 16 + ln];
    wv0.x = Wv[ka * FEATD + ln];        wv0.y = Wv[(ka + 1) * FEATD + ln];
    wv1.x = Wv[ka * FEATD + 16 + ln];   wv1.y = Wv[(ka + 1) * FEATD + 16 + ln];
    au0 = __builtin_amdgcn_wmma_f32_16x16x4_f32(false, a, false, wu0, (short)0, au0, false, false);
    au1 = __builtin_amdgcn_wmma_f32_16x16x4_f32(false, a, false, wu1, (short)0, au1, false, false);
    av0 = __builtin_amdgcn_wmma_f32_16x16x4_f32(false, a, false, wv0, (short)0, av0, false, false);
    av1 = __builtin_amdgcn_wmma_f32_16x16x4_f32(false, a, false, wv1, (short)0, av1, false, false);
  }
  #pragma unroll
  for (int i = 0; i < 8; ++i) {
    int m = row0 + i + 8 * kh;
    u[m * FEATD + ln]      = au0[i];
    u[m * FEATD + 16 + ln] = au1[i];
    v[m * FEATD + ln]      = av0[i];
    v[m * FEATD + 16 + ln] = av1[i];
  }
}

// ---------------------------------------------------------------------------
// Kernel 3: layer-1 aggregation. One wave per node, lane = feature channel.
// out[n,f] = mean_a relu( v[dst[eid]]f + u[src[eid]]f * sp_dist[eid] )
// Edge messages are never materialized; u/v/sp_dist/src/dst are L2-resident.
// ---------------------------------------------------------------------------
__global__ void pgnn_aggregate(const float* __restrict__ u,
                               const float* __restrict__ v,
                               const float* __restrict__ sp_dist,
                               const int* __restrict__ src,
                               const int* __restrict__ dst,
                               const int* __restrict__ anchor_eid,
                               float* __restrict__ out) {
  int node = blockIdx.x * (blockDim.x >> 5) + (threadIdx.x >> 5);
  if (node >= NNODES) return;
  int lane = threadIdx.x & 31;
  const int* ae = anchor_eid + (size_t)node * NANCH;
  float acc = 0.0f;
  for (int a = 0; a < NANCH; ++a) {
    int eid = __builtin_amdgcn_readfirstlane(ae[a]);   // wave-uniform -> SGPR
    int s = __builtin_amdgcn_readfirstlane(src[eid]);
    int d = __builtin_amdgcn_readfirstlane(dst[eid]);
    float w = sp_dist[eid];
    float m = v[(size_t)d * FEATD + lane] + u[(size_t)s * FEATD + lane] * w;
    acc += fmaxf(m, 0.0f);
  }
  out[(size_t)node * FEATD + lane] = acc * (1.0f / NANCH);
}

// ---------------------------------------------------------------------------
// Kernel 4: layer-2 position head + L2 normalize, fused. One wave per node.
// For each anchor: dot32(relu(msg), Wo2) + bo2 via wave32 butterfly; lane a&31
// keeps pos for anchors {lane, lane+32}; then one more butterfly for the norm.
// ---------------------------------------------------------------------------
__global__ void pgnn_position(const float* __restrict__ u,
                              const float* __restrict__ v,
                              const float* __restrict__ sp_dist,
                              const int* __restrict__ src,
                              const int* __restrict__ dst,
                              const int* __restrict__ anchor_eid,
                              const float* __restrict__ Wo,
                              const float* __restrict__ bo,
                              float* __restrict__ out) {
  int node = blockIdx.x * (blockDim.x >> 5) + (threadIdx.x >> 5);
  if (node >= NNODES) return;
  int lane = threadIdx.x & 31;
  const int* ae = anchor_eid + (size_t)node * NANCH;
  float wo = Wo[lane];                     // Wo2 is [32,1]
  float bias = bo[0];
  float p0 = 0.0f, p1 = 0.0f;
  for (int a = 0; a < NANCH; ++a) {
    int eid = __builtin_amdgcn_readfirstlane(ae[a]);
    int s = __builtin_amdgcn_readfirstlane(src[eid]);
    int d = __builtin_amdgcn_readfirstlane(dst[eid]);
    float w = sp_dist[eid];
    float m = fmaxf(v[(size_t)d * FEATD + lane] + u[(size_t)s * FEATD + lane] * w, 0.0f);
    float t = m * wo;
    #pragma unroll
    for (int off = 16; off >= 1; off >>= 1) t += __shfl_xor(t, off, 32);
    t += bias;                             // all lanes hold the dot now
    if (lane == (a & 31)) { if (a < 32) p0 = t; else p1 = t; }
  }
  float ss = p0 * p0 + p1 * p1;
  #pragma unroll
  for (int off = 16; off >= 1; off >>= 1) ss += __shfl_xor(ss, off, 32);
  float inv = 1.0f / fmaxf(sqrtf(ss), 1e-12f);
  out[(size_t)node * NANCH + lane]      = p0 * inv;
  out[(size_t)node * NANCH + 32 + lane] = p1 * inv;
}

// ---------------------------------------------------------------------------
// Launch: pre_gemm -> uv_gemm(L1) -> aggregate -> uv_gemm(L2) -> position
// Workspace layout (floats): [x | u | v], x buffer reused for the layer-1
// aggregate output (x1 is dead by then), u/v overwritten for layer 2.
// ---------------------------------------------------------------------------
extern "C" void kernel_launch(void* const* d_in, const int* in_sizes, int n_in,
                              void* d_out, int out_size, void* d_ws, size_t ws_size,
                              hipStream_t stream) {
  const float* feat       = (const float*)d_in[0];
  const float* sp_dist    = (const float*)d_in[1];
  const int*   src        = (const int*)d_in[2];
  const int*   dst        = (const int*)d_in[3];
  const int*   anchor_eid = (const int*)d_in[4];
  // d_in[5] = dists_max (unused, shape only)
  const float* W_pre = (const float*)d_in[6];
  const float* b_pre = (const float*)d_in[7];
  const float* Wu1   = (const float*)d_in[8];
  const float* bu1   = (const float*)d_in[9];
  const float* Wv1   = (const float*)d_in[10];
  const float* bv1   = (const float*)d_in[11];
  // d_in[12], d_in[13] = Wo1/bo1 (layer-1 position head is discarded)
  const float* Wu2   = (const float*)d_in[14];
  const float* bu2   = (const float*)d_in[15];
  const float* Wv2   = (const float*)d_in[16];
  const float* bv2   = (const float*)d_in[17];
  const float* Wo2   = (const float*)d_in[18];
  const float* bo2   = (const float*)d_in[19];
  float* out = (float*)d_out;

  const size_t NX = (size_t)NNODES * FEATD;       // 1.6M floats = 6.4 MB
  float* x = (float*)d_ws;
  float* u = x + NX;
  float* v = u + NX;

  const int n_tiles = NNODES / 16;                // 3125, exact
  const int gemm_blocks = (n_tiles + 7) / 8;      // 8 waves / 256-thread block
  const int node_blocks = NNODES / 8;             // 6250, exact

  pgnn_pre_gemm<<<gemm_blocks, 256, 0, stream>>>(feat, W_pre, b_pre, x, n_tiles);
  pgnn_uv_gemm<<<gemm_blocks, 256, 0, stream>>>(x, Wu1, bu1, Wv1, bv1, u, v, n_tiles);
  pgnn_aggregate<<<node_blocks, 256, 0, stream>>>(u, v, sp_dist, src, dst, anchor_eid, x);
  pgnn_uv_gemm<<<gemm_blocks, 256, 0, stream>>>(x, Wu2, bu2, Wv2, bv2, u, v, n_tiles);
  pgnn_position<<<node_blocks, 256, 0, stream>>>(u, v, sp_dist, src, dst, anchor_eid,
                                                 Wo2, bo2, out);
}